// CustomLoss_188978561648
// MI455X (gfx1250) — compile-verified
//
#include <hip/hip_runtime.h>

typedef __attribute__((ext_vector_type(2))) float v2f;
typedef __attribute__((ext_vector_type(8))) float v8f;

#define KROWS  64
#define NDIM   4096
#define NB     64                 // n-chunk staged in LDS per iteration
#define LDSW   (NB + 4)           // padded row stride (floats): conflict-free, 16B aligned
#define NCHUNK (NDIM / NB)        // 64 chunks

__global__ void zero_out_kernel(float* __restrict__ out) {
    if (threadIdx.x == 0) out[0] = 0.0f;
}

// CDNA5 async copy: global -> LDS, 16B per lane, tracked by ASYNCcnt.
// VGLOBAL encoding: VDST = LDS byte address VGPR, VADDR = 64-bit global address.
__device__ __forceinline__ void async_load_b128(unsigned lds_addr, const float* gaddr) {
    asm volatile("global_load_async_to_lds_b128 %0, %1, off"
                 :: "v"(lds_addr), "v"((unsigned long long)(uintptr_t)gaddr)
                 : "memory");
}

// D[k,i] = sum_n zs[k,n] * X[i,n]; loss = (1/K) * sum_{k,i} -num/(vn + D - num),
// num = zs[k,i]*X[i,i].
__global__ __launch_bounds__(128) void ee_loss_wmma_f32_kernel(
    const float* __restrict__ zs,          // [64, 4096]
    const float* __restrict__ X,           // [4096, 4096]
    const float* __restrict__ var_noise,   // [1]
    float* __restrict__ out)               // [1]
{
    __shared__ float Xs [2][16 * LDSW];    // X  tile: 16 i-rows x NB n-cols, double buffered
    __shared__ float Zss[2][64 * LDSW];    // zs tile: 64 k-rows x NB n-cols, double buffered

    const int tid  = threadIdx.x;
    const int wave = tid >> 5;             // 0..3  -> k-tile (rows 16*wave..)
    const int lane = tid & 31;
    const int l16  = lane & 15;
    const int half = lane >> 4;            // 0: K=0,1 / M=v ; 1: K=2,3 / M=v+8
    const int i0   = blockIdx.x * 16;      // i-tile base (256 blocks cover 4096)

    v8f acc = {};                          // 16x16 f32 accumulator tile

    // Issue one chunk's worth of async global->LDS copies (10 ops per thread):
    //   X tile : 16x64 floats = 256 float4  -> 2 per thread
    //   zs tile: 64x64 floats = 1024 float4 -> 8 per thread
    auto issue_chunk = [&](int n0, int buf) {
        #pragma unroll
        for (int j = 0; j < 2; ++j) {
            int fl = tid + 128 * j;        // 0..255
            int r  = fl >> 4, c4 = fl & 15;
            async_load_b128((unsigned)(uintptr_t)&Xs[buf][r * LDSW + c4 * 4],
                            X + (size_t)(i0 + r) * NDIM + n0 + c4 * 4);
        }
        #pragma unroll
        for (int j = 0; j < 8; ++j) {
            int fl = tid + 128 * j;        // 0..1023
            int r  = fl >> 4, c4 = fl & 15;
            async_load_b128((unsigned)(uintptr_t)&Zss[buf][r * LDSW + c4 * 4],
                            zs + (size_t)r * NDIM + n0 + c4 * 4);
        }
    };

    issue_chunk(0, 0);                     // prologue: chunk 0 -> buffer 0

    int buf = 0;
    for (int t = 0; t < NCHUNK; ++t) {
        if (t + 1 < NCHUNK) {
            issue_chunk((t + 1) * NB, buf ^ 1);       // prefetch next chunk
            // outstanding <= 20; oldest 10 (current buffer) must retire
            asm volatile("s_wait_asynccnt 0xa" ::: "memory");
        } else {
            asm volatile("s_wait_asynccnt 0x0" ::: "memory");
        }
        __syncthreads();                   // all waves' current-buffer data in LDS

        // A fragment: lane holds zs[16*wave + l16, n + half*2 + {0,1}]
        // B fragment: lane holds X [i0 + l16,      n + half*2 + {0,1}]
        const float* arow = &Zss[buf][(wave * 16 + l16) * LDSW];
        const float* brow = &Xs [buf][l16 * LDSW];

        #pragma unroll
        for (int nl = 0; nl < NB; nl += 4) {
            v2f a = *(const v2f*)(arow + nl + half * 2);
            v2f b = *(const v2f*)(brow + nl + half * 2);
            acc = __builtin_amdgcn_wmma_f32_16x16x4_f32(
                /*neg_a=*/false, a, /*neg_b=*/false, b,
                /*c_mod=*/(short)0, acc, /*reuse_a=*/false, /*reuse_b=*/false);
        }
        __syncthreads();                   // everyone done reading before overwrite
        buf ^= 1;
    }

    // ---- epilogue: C layout is VGPR v, lane L -> M = v + 8*half, N = l16 ----
    const int   i     = i0 + l16;
    const float xdiag = X[(size_t)i * NDIM + i];
    const float vn    = var_noise[0];

    float s = 0.0f;
    #pragma unroll
    for (int v = 0; v < 8; ++v) {
        int   k   = wave * 16 + v + 8 * half;
        float num = zs[(size_t)k * NDIM + i] * xdiag;
        float den = acc[v] - num;
        s -= num / (vn + den);
    }

    // wave32 reduction
    #pragma unroll
    for (int off = 16; off > 0; off >>= 1)
        s += __shfl_down(s, off, 32);

    if (lane == 0)
        atomicAdd(out, s * (1.0f / (float)KROWS));
}

extern "C" void kernel_launch(void* const* d_in, const int* in_sizes, int n_in,
                              void* d_out, int out_size, void* d_ws, size_t ws_size,
                              hipStream_t stream) {
    const float* zs = (const float*)d_in[0];   // [64*4096]
    const float* X  = (const float*)d_in[1];   // [4096*4096]
    const float* vn = (const float*)d_in[2];   // [1]
    float* out = (float*)d_out;

    hipLaunchKernelGGL(zero_out_kernel, dim3(1), dim3(32), 0, stream, out);
    hipLaunchKernelGGL(ee_loss_wmma_f32_kernel, dim3(NDIM / 16), dim3(128), 0, stream,
                       zs, X, vn, out);
}